// SSNN_32590211842465
// MI455X (gfx1250) — compile-verified
//
#include <hip/hip_runtime.h>
#include <math.h>

// ---------------------------------------------------------------------------
// Linear RNN (SSM) via chunked parallel scan, fp32 WMMA (gfx1250).
//
//   x[n+1] = A x[n] + B u[n],  y[n] = tanh(C x[n+1] + D u[n]),  out = y^T
//   N = 16384, X = 1024, U = Y = 256.
//
// Pipeline (all on `stream`, graph-capture safe):
//   0. transpose B, C, D (LDS tiled)           -> Bt, Ct, Dt
//   1. Bu = u @ Bt                 (WMMA GEMM 64x64 wave tile)
//   2. M = A^128 by 7 squarings    (WMMA GEMM 32x32 wave tile, 1024^3 each)
//   3. pass1: 128 serial steps  S = A@S + Bu_t   (WMMA 16x16 tiles, 4 K-chains
//             for intra-wave ILP; 128 chunks in parallel)
//   4. pass2: 127 serial matvecs  e_j = M e_{j-1} + send_{j-1}
//   5. pass3: rerun scan seeded with true chunk starts, store xs
//   6. out = tanh(xs@Ct + u@Dt), stored transposed (WMMA 64x64 + tanh)
//
// Working set ~147 MB => resident in MI455X 192 MB L2; this problem is
// bounded by serial-scan latency + fp32 matrix throughput, not HBM.
// On the serial path each wave has ~1 wave/SIMD, so dependent-WMMA latency
// must be hidden *inside* the wave: 4 independent accumulator chains.
// ---------------------------------------------------------------------------

typedef __attribute__((ext_vector_type(2))) float v2f;
typedef __attribute__((ext_vector_type(8))) float v8f;

#define NSEQ   16384
#define XL     1024
#define UL     256
#define YLN    256
#define CHUNKS 128
#define CLEN   128
#define COLSTR ((size_t)CLEN * XL)   // 131072: element (x,j) of a time-slot

__device__ __forceinline__ v8f wmma4(v2f a, v2f b, v8f c) {
  // V_WMMA_F32_16X16X4_F32 : D(16x16,f32) = A(16x4) * B(4x16) + C
  return __builtin_amdgcn_wmma_f32_16x16x4_f32(
      /*neg_a=*/false, a, /*neg_b=*/false, b,
      /*c_mod=*/(short)0, c, /*reuse_a=*/false, /*reuse_b=*/false);
}

// ---------------------------------------------------------------------------
// Generic C = A(MxK,row) @ B(KxN,row).  Per-wave tile = (16*WM) x (16*WN)
// WMMA tiles; workgroup = 8 waves as 2(m) x 4(n).  Dims must divide tiles.
// ---------------------------------------------------------------------------
template <int WM, int WN>
__global__ __launch_bounds__(256) void gemm_rr_k(
    const float* __restrict__ Ap, const float* __restrict__ Bp,
    float* __restrict__ Cp, int K, int lda, int ldb, int ldc) {
  const int lane = threadIdx.x & 31;
  const int w    = threadIdx.x >> 5;
  const int wm = w >> 2, wn = w & 3;
  const int tm = blockIdx.x * (32 * WM) + wm * (16 * WM);
  const int tn = blockIdx.y * (64 * WN) + wn * (16 * WN);
  const int ml = lane & 15;            // M (or N) index inside 16-tile
  const int kh = (lane >> 4) << 1;     // K pair: lanes 0-15 -> k0,k0+1; 16-31 -> k0+2,k0+3

  v8f acc[WM][WN];
#pragma unroll
  for (int mi = 0; mi < WM; ++mi)
#pragma unroll
    for (int ni = 0; ni < WN; ++ni) acc[mi][ni] = (v8f){};

  const float* ap[WM];
  const float* bp[WN];
#pragma unroll
  for (int mi = 0; mi < WM; ++mi)
    ap[mi] = Ap + (size_t)(tm + mi * 16 + ml) * lda + kh;
#pragma unroll
  for (int ni = 0; ni < WN; ++ni)
    bp[ni] = Bp + (size_t)kh * ldb + (tn + ni * 16 + ml);

  for (int k = 0; k < K; k += 4) {
    v2f af[WM], bf[WN];
    const size_t bo0 = (size_t)k * ldb, bo1 = bo0 + (size_t)ldb;
#pragma unroll
    for (int mi = 0; mi < WM; ++mi) af[mi] = (v2f){ ap[mi][k], ap[mi][k + 1] };
#pragma unroll
    for (int ni = 0; ni < WN; ++ni) bf[ni] = (v2f){ bp[ni][bo0], bp[ni][bo1] };
#pragma unroll
    for (int mi = 0; mi < WM; ++mi)
#pragma unroll
      for (int ni = 0; ni < WN; ++ni)
        acc[mi][ni] = wmma4(af[mi], bf[ni], acc[mi][ni]);
  }

  const int nn = lane & 15;
  const int mb = (lane >> 4) << 3;     // rows r..r+7 are M = mb+r
#pragma unroll
  for (int mi = 0; mi < WM; ++mi)
#pragma unroll
    for (int ni = 0; ni < WN; ++ni) {
      float* cp = Cp + (size_t)(tm + mi * 16 + mb) * ldc + (tn + ni * 16 + nn);
#pragma unroll
      for (int r = 0; r < 8; ++r) cp[(size_t)r * ldc] = acc[mi][ni][r];
    }
}

// ---------------------------------------------------------------------------
// One serial scan step over all chunks in parallel:
//   Snew(XL x CHUNKS) = A(XL x XL) @ Sold + Bu_slot   [+ optional xs store]
// S matrices row-major with ld = CHUNKS.  addend/xs element (x,j) lives at
// base[j*COLSTR + x]  (== Bu[(j*CLEN + t)*XL + x]).
// Wave tile 16x16, K=1024 split into 4 independent accumulator chains
// (intra-wave WMMA ILP: the serial path has ~1 wave/SIMD, so cross-wave
// latency hiding is unavailable).  WG = 8 waves as 2(m) x 4(n) -> 32x64;
// grid (32, 2) = 64 WGs / 512 waves, one 16x16 tile per wave.
// ---------------------------------------------------------------------------
template <bool STORE_XS>
__global__ __launch_bounds__(256) void scan_step_k(
    const float* __restrict__ Ap, const float* __restrict__ Sold,
    const float* __restrict__ addb, float* __restrict__ Snew,
    float* __restrict__ xsb) {
  const int lane = threadIdx.x & 31;
  const int w    = threadIdx.x >> 5;
  const int wm = w >> 2, wn = w & 3;
  const int tm = blockIdx.x * 32 + wm * 16;
  const int tn = blockIdx.y * 64 + wn * 16;
  const int ml = lane & 15;
  const int kh = (lane >> 4) << 1;

  constexpr int KQ = XL / 4;                    // 256 per chain
  constexpr size_t BQ = (size_t)KQ * CHUNKS;    // B-operand chain stride

  v8f acc0 = {}, acc1 = {}, acc2 = {}, acc3 = {};
  const float* ap = Ap + (size_t)(tm + ml) * XL + kh;
  const float* bp = Sold + (size_t)kh * CHUNKS + (tn + ml);

  for (int k = 0; k < KQ; k += 4) {
    v2f a0 = { ap[k],          ap[k + 1] };
    v2f a1 = { ap[k + KQ],     ap[k + KQ + 1] };
    v2f a2 = { ap[k + 2 * KQ], ap[k + 2 * KQ + 1] };
    v2f a3 = { ap[k + 3 * KQ], ap[k + 3 * KQ + 1] };
    const size_t bo = (size_t)k * CHUNKS;
    v2f b0 = { bp[bo],          bp[bo + CHUNKS] };
    v2f b1 = { bp[bo + BQ],     bp[bo + BQ + CHUNKS] };
    v2f b2 = { bp[bo + 2 * BQ], bp[bo + 2 * BQ + CHUNKS] };
    v2f b3 = { bp[bo + 3 * BQ], bp[bo + 3 * BQ + CHUNKS] };
    acc0 = wmma4(a0, b0, acc0);
    acc1 = wmma4(a1, b1, acc1);
    acc2 = wmma4(a2, b2, acc2);
    acc3 = wmma4(a3, b3, acc3);
  }
  v8f acc = (acc0 + acc1) + (acc2 + acc3);

  const int j  = tn + (lane & 15);
  const int mb = tm + ((lane >> 4) << 3);
  const float* addc = addb + (size_t)j * COLSTR + mb;
  if constexpr (STORE_XS) {
    float* xsc = xsb + (size_t)j * COLSTR + mb;
#pragma unroll
    for (int r = 0; r < 8; ++r) {
      float v = acc[r] + addc[r];
      Snew[(size_t)(mb + r) * CHUNKS + j] = v;
      xsc[r] = v;
    }
  } else {
    (void)xsb;
#pragma unroll
    for (int r = 0; r < 8; ++r) {
      float v = acc[r] + addc[r];
      Snew[(size_t)(mb + r) * CHUNKS + j] = v;
    }
  }
}

// ---------------------------------------------------------------------------
// Serial chunk-boundary propagation:  vout = M @ vin + vadd   (all XL long,
// strided columns of row-major XL x CHUNKS buffers).  Grid 32 x 256 threads:
// 256 waves, 4 rows each; v staged in LDS; wave-shuffle reduction (wave32).
// ---------------------------------------------------------------------------
__global__ __launch_bounds__(256) void matvec_step_k(
    const float* __restrict__ Mm, const float* __restrict__ vin, int vins,
    const float* __restrict__ vadd, int vadds,
    float* __restrict__ vout, int vouts) {
  __shared__ float vsh[XL];
  const int tid = threadIdx.x;
  for (int i = tid; i < XL; i += 256) vsh[i] = vin[(size_t)i * vins];
  __syncthreads();

  const int lane = tid & 31, wid = tid >> 5;
  const int gw = blockIdx.x * 8 + wid;          // 0..255
#pragma unroll
  for (int rr = 0; rr < 4; ++rr) {
    const int row = gw * 4 + rr;
    const float* Mr = Mm + (size_t)row * XL;
    float s = 0.f;
    for (int i = 0; i < XL; i += 32) s += Mr[i + lane] * vsh[i + lane];
#pragma unroll
    for (int off = 16; off > 0; off >>= 1) s += __shfl_down(s, off, 32);
    if (lane == 0) vout[(size_t)row * vouts] = s + vadd[(size_t)row * vadds];
  }
}

// ---------------------------------------------------------------------------
// out[yl, n] = tanh( xs(N x XL)@Ct(XL x YL) + u(N x UL)@Dt(UL x YL) ),
// stored transposed: out[yl*NSEQ + n].  64x64 wave tile (4x4 WMMA tiles),
// WG = 8 waves as 2(m) x 4(n) -> 128x256; grid (128, 1).
// ---------------------------------------------------------------------------
__global__ __launch_bounds__(256) void final_gemm_k(
    const float* __restrict__ xs, const float* __restrict__ Ct,
    const float* __restrict__ uf, const float* __restrict__ Dt,
    float* __restrict__ out) {
  const int lane = threadIdx.x & 31;
  const int w    = threadIdx.x >> 5;
  const int wm = w >> 2, wn = w & 3;
  const int tm = blockIdx.x * 128 + wm * 64;    // sequence index n
  const int tn = wn * 64;                       // output feature yl (N = 256)
  const int ml = lane & 15;
  const int kh = (lane >> 4) << 1;

  v8f acc[4][4];
#pragma unroll
  for (int mi = 0; mi < 4; ++mi)
#pragma unroll
    for (int ni = 0; ni < 4; ++ni) acc[mi][ni] = (v8f){};

  {  // xs @ Ct, K = XL
    const float* ap[4];
    const float* bp[4];
#pragma unroll
    for (int mi = 0; mi < 4; ++mi)
      ap[mi] = xs + (size_t)(tm + mi * 16 + ml) * XL + kh;
#pragma unroll
    for (int ni = 0; ni < 4; ++ni)
      bp[ni] = Ct + (size_t)kh * YLN + (tn + ni * 16 + ml);
    for (int k = 0; k < XL; k += 4) {
      v2f af[4], bf[4];
      const size_t bo0 = (size_t)k * YLN, bo1 = bo0 + YLN;
#pragma unroll
      for (int mi = 0; mi < 4; ++mi) af[mi] = (v2f){ ap[mi][k], ap[mi][k + 1] };
#pragma unroll
      for (int ni = 0; ni < 4; ++ni) bf[ni] = (v2f){ bp[ni][bo0], bp[ni][bo1] };
#pragma unroll
      for (int mi = 0; mi < 4; ++mi)
#pragma unroll
        for (int ni = 0; ni < 4; ++ni)
          acc[mi][ni] = wmma4(af[mi], bf[ni], acc[mi][ni]);
    }
  }
  {  // u @ Dt, K = UL
    const float* ap[4];
    const float* bp[4];
#pragma unroll
    for (int mi = 0; mi < 4; ++mi)
      ap[mi] = uf + (size_t)(tm + mi * 16 + ml) * UL + kh;
#pragma unroll
    for (int ni = 0; ni < 4; ++ni)
      bp[ni] = Dt + (size_t)kh * YLN + (tn + ni * 16 + ml);
    for (int k = 0; k < UL; k += 4) {
      v2f af[4], bf[4];
      const size_t bo0 = (size_t)k * YLN, bo1 = bo0 + YLN;
#pragma unroll
      for (int mi = 0; mi < 4; ++mi) af[mi] = (v2f){ ap[mi][k], ap[mi][k + 1] };
#pragma unroll
      for (int ni = 0; ni < 4; ++ni) bf[ni] = (v2f){ bp[ni][bo0], bp[ni][bo1] };
#pragma unroll
      for (int mi = 0; mi < 4; ++mi)
#pragma unroll
        for (int ni = 0; ni < 4; ++ni)
          acc[mi][ni] = wmma4(af[mi], bf[ni], acc[mi][ni]);
    }
  }

  const int nn = lane & 15;
  const int mb = (lane >> 4) << 3;
#pragma unroll
  for (int mi = 0; mi < 4; ++mi)
#pragma unroll
    for (int ni = 0; ni < 4; ++ni) {
      const int yl = tn + ni * 16 + nn;
      const int n0 = tm + mi * 16 + mb;
      float* op = out + (size_t)yl * NSEQ + n0;   // 8 consecutive floats/lane
#pragma unroll
      for (int r = 0; r < 8; ++r) op[r] = tanhf(acc[mi][ni][r]);
    }
}

// ---------------------------------------------------------------------------
// LDS-tiled transpose: out[c*rows + r] = in[r*cols + c].  Block (32,8).
// ---------------------------------------------------------------------------
__global__ __launch_bounds__(256) void transpose_k(
    const float* __restrict__ in, float* __restrict__ out, int rows, int cols) {
  __shared__ float tile[32][33];
  const int c0 = blockIdx.x * 32, r0 = blockIdx.y * 32;
  const int tx = threadIdx.x, ty = threadIdx.y;
#pragma unroll
  for (int i = 0; i < 32; i += 8)
    tile[ty + i][tx] = in[(size_t)(r0 + ty + i) * cols + (c0 + tx)];
  __syncthreads();
#pragma unroll
  for (int i = 0; i < 32; i += 8)
    out[(size_t)(c0 + ty + i) * rows + (r0 + tx)] = tile[tx][ty + i];
}

__global__ void set_col_k(float* __restrict__ dst, const float* __restrict__ src,
                          int stride, int n) {
  int i = blockIdx.x * blockDim.x + threadIdx.x;
  if (i < n) dst[(size_t)i * stride] = src[i];
}

// ---------------------------------------------------------------------------
extern "C" void kernel_launch(void* const* d_in, const int* in_sizes, int n_in,
                              void* d_out, int out_size, void* d_ws, size_t ws_size,
                              hipStream_t stream) {
  (void)in_sizes; (void)n_in; (void)out_size; (void)ws_size;
  const float* u  = (const float*)d_in[0];   // (16384, 256) row-major
  const float* A  = (const float*)d_in[1];   // (1024, 1024)
  const float* B  = (const float*)d_in[2];   // (1024, 256)
  const float* C  = (const float*)d_in[3];   // (256, 1024)
  const float* D  = (const float*)d_in[4];   // (256, 256)
  const float* x0 = (const float*)d_in[5];   // (1024,)
  float* out = (float*)d_out;                // (256, 16384)

  float* ws = (float*)d_ws;
  size_t off = 0;
  float* Bu = ws + off; off += (size_t)NSEQ * XL;     // 64 MB
  float* xs = ws + off; off += (size_t)NSEQ * XL;     // 64 MB
  float* Bt = ws + off; off += (size_t)UL * XL;       // B^T  (256 x 1024)
  float* Ct = ws + off; off += (size_t)XL * YLN;      // C^T  (1024 x 256)
  float* Dt = ws + off; off += (size_t)UL * YLN;      // D^T  (256 x 256)
  float* Q0 = ws + off; off += (size_t)XL * XL;       // A-power ping
  float* Q1 = ws + off; off += (size_t)XL * XL;       // A-power pong
  float* SA = ws + off; off += (size_t)XL * CHUNKS;   // scan ping
  float* SB = ws + off; off += (size_t)XL * CHUNKS;   // scan pong
  float* EB = ws + off; off += (size_t)XL * CHUNKS;   // chunk-start states

  // 0) transposes
  transpose_k<<<dim3(UL / 32, XL / 32), dim3(32, 8), 0, stream>>>(B, Bt, XL, UL);
  transpose_k<<<dim3(XL / 32, YLN / 32), dim3(32, 8), 0, stream>>>(C, Ct, YLN, XL);
  transpose_k<<<dim3(YLN / 32, UL / 32), dim3(32, 8), 0, stream>>>(D, Dt, UL, YLN);

  // 1) Bu = u @ B^T  (16384 x 1024 x 256), 64x64 wave tiles -> 512 WGs
  gemm_rr_k<4, 4><<<dim3(NSEQ / 128, XL / 256), 256, 0, stream>>>(
      u, Bt, Bu, UL, UL, XL, XL);

  // 2) M = A^128 by repeated squaring (7 products, ends in Q0)
  //    32x32 wave tiles keep 128 WGs in flight on the small 1024^2 grid.
  gemm_rr_k<2, 2><<<dim3(XL / 64, XL / 128), 256, 0, stream>>>(
      A, A, Q0, XL, XL, XL, XL);
  {
    float* src = Q0; float* dst = Q1;
    for (int i = 0; i < 6; ++i) {
      gemm_rr_k<2, 2><<<dim3(XL / 64, XL / 128), 256, 0, stream>>>(
          src, src, dst, XL, XL, XL, XL);
      float* t = src; src = dst; dst = t;
    }
  }
  const float* M128 = Q0;

  // init: pass-1 state = 0; chunk-start col 0 = x0
  hipMemsetAsync(SA, 0, (size_t)XL * CHUNKS * sizeof(float), stream);
  hipMemsetAsync(EB, 0, (size_t)XL * CHUNKS * sizeof(float), stream);
  set_col_k<<<4, 256, 0, stream>>>(EB, x0, CHUNKS, XL);

  // 3) pass 1: local scans (zero init), 128 chunks in parallel
  {
    float* cur = SA; float* nxt = SB;
    for (int t = 0; t < CLEN; ++t) {
      scan_step_k<false><<<dim3(XL / 32, CHUNKS / 64), 256, 0, stream>>>(
          A, cur, Bu + (size_t)t * XL, nxt, nullptr);
      float* tmp = cur; cur = nxt; nxt = tmp;
    }
    // after 128 steps cur == SA holds per-chunk local end states (send_j)
  }

  // 4) pass 2: e_j = A^128 e_{j-1} + send_{j-1}   (127 serial matvecs)
  for (int j = 1; j < CHUNKS; ++j) {
    matvec_step_k<<<32, 256, 0, stream>>>(M128, EB + (j - 1), CHUNKS,
                                          SA + (j - 1), CHUNKS, EB + j, CHUNKS);
  }

  // 5) pass 3: rerun scans seeded with true chunk starts, store xs
  {
    float* cur = EB; float* nxt = SB;
    for (int t = 0; t < CLEN; ++t) {
      scan_step_k<true><<<dim3(XL / 32, CHUNKS / 64), 256, 0, stream>>>(
          A, cur, Bu + (size_t)t * XL, nxt, xs + (size_t)t * XL);
      float* tmp = cur; cur = nxt; nxt = tmp;
    }
  }

  // 6) out = tanh(xs @ C^T + u @ D^T), transposed store
  final_gemm_k<<<dim3(NSEQ / 128, 1), 256, 0, stream>>>(xs, Ct, u, Dt, out);
}